// VirtualNormalLoss_38843684225117
// MI455X (gfx1250) — compile-verified
//
#include <hip/hip_runtime.h>

// Problem constants (match reference)
#define BATCH   64
#define HH      480
#define WW      640
#define HW      (HH * WW)
#define NG      100000
#define SPI     3
#define G3      (NG * SPI)
#define TPB     256
#define INV_F   (1.0f / 518.857f)   // focal at W=640
#define CX      320.0f
#define CY      240.0f

__device__ __forceinline__ void tri_normal(const float* X, const float* Y,
                                           const float* z,
                                           float& nx, float& ny, float& nz) {
    // points p_k = (X_k*z_k, Y_k*z_k, z_k); a = p1-p0, b = p2-p0, n = a x b
    float ax = X[1] * z[1] - X[0] * z[0];
    float ay = Y[1] * z[1] - Y[0] * z[0];
    float az = z[1] - z[0];
    float bx = X[2] * z[2] - X[0] * z[0];
    float by = Y[2] * z[2] - Y[0] * z[0];
    float bz = z[2] - z[0];
    nx = ay * bz - az * by;
    ny = az * bx - ax * bz;
    nz = ax * by - ay * bx;
}

__global__ __launch_bounds__(TPB) void vnl_partials_kernel(
    const float* __restrict__ pred, const float* __restrict__ gt,
    const int* __restrict__ idx, float* __restrict__ partials) {
    __shared__ int   s_idx[TPB * SPI];   // 3072 B: this block's idx slice
    __shared__ float s_red[TPB / 32];

    const int tid = threadIdx.x;
    const int b   = blockIdx.y;
    const int gi0 = blockIdx.x * TPB;

    // ---- Stage idx[b, 3*gi0 : 3*(gi0+TPB)] into LDS via async DMA (CDNA5) ----
    // 768 ints = 3072 B = 192 x B128 chunks; lanes past the valid range clamp
    // their source to the row base so the last block never reads past the
    // buffer (valid bytes = valid_groups*12, a multiple of 16 here: 160*12).
    {
        const int valid_groups = (NG - gi0) < TPB ? (NG - gi0) : TPB;
        const int valid_bytes  = valid_groups * 12;
        const char* row = (const char*)(idx + (size_t)b * G3 + (size_t)gi0 * SPI);
        if (tid < (TPB * SPI * 4) / 16) {            // 192 lanes
            int byte_off = tid * 16;
            unsigned long long gaddr =
                (unsigned long long)(row + ((byte_off + 16 <= valid_bytes) ? byte_off : 0));
            unsigned lds_addr =
                (unsigned)(unsigned long long)(&s_idx[0]) + (unsigned)byte_off;
            asm volatile("global_load_async_to_lds_b128 %0, %1, off"
                         :: "v"(lds_addr), "v"(gaddr)
                         : "memory");
        }
        asm volatile("s_wait_asynccnt 0" ::: "memory");
        __syncthreads();
    }

    // ---- Per-thread loss for one triangle group ----
    float s = 0.0f;
    const int gi = gi0 + tid;
    if (gi < NG) {
        const size_t base = (size_t)b * HW;
        float X[3], Y[3], zp[3], zg[3];
#pragma unroll
        for (int k = 0; k < SPI; ++k) {
            unsigned i = (unsigned)s_idx[tid * SPI + k];
            unsigned y = i / (unsigned)WW;            // magic-number div
            unsigned x = i - y * (unsigned)WW;
            X[k]  = ((float)x - CX) * INV_F;
            Y[k]  = ((float)y - CY) * INV_F;
            zp[k] = pred[base + i];                   // random gather (L2-resident)
            zg[k] = gt[base + i];
        }
        float npx, npy, npz, ngx, ngy, ngz;
        tri_normal(X, Y, zp, npx, npy, npz);
        tri_normal(X, Y, zg, ngx, ngy, ngz);
        float dp = npx * ngx + npy * ngy + npz * ngz;
        float lp = sqrtf(npx * npx + npy * npy + npz * npz) + 1e-8f;
        float lg = sqrtf(ngx * ngx + ngy * ngy + ngz * ngz) + 1e-8f;
        float c  = dp / (lp * lg);
        c = fminf(1.0f, fmaxf(-1.0f, c));
        s = 1.0f - c;
    }

    // ---- wave32 reduce, then cross-wave via LDS (deterministic) ----
#pragma unroll
    for (int off = 16; off > 0; off >>= 1)
        s += __shfl_down(s, off, 32);
    if ((tid & 31) == 0) s_red[tid >> 5] = s;
    __syncthreads();
    if (tid == 0) {
        float t = 0.0f;
#pragma unroll
        for (int w = 0; w < TPB / 32; ++w) t += s_red[w];
        partials[(size_t)b * gridDim.x + blockIdx.x] = t;
    }
}

__global__ __launch_bounds__(TPB) void vnl_finalize_kernel(
    const float* __restrict__ partials, int n, float* __restrict__ out) {
    __shared__ float sm[TPB];
    const int tid = threadIdx.x;
    float s = 0.0f;
    for (int i = tid; i < n; i += TPB) s += partials[i];
    sm[tid] = s;
    __syncthreads();
#pragma unroll
    for (int stride = TPB / 2; stride > 0; stride >>= 1) {
        if (tid < stride) sm[tid] += sm[tid + stride];
        __syncthreads();
    }
    if (tid == 0)
        out[0] = sm[0] * (1.0f / ((float)BATCH * (float)NG));
}

extern "C" void kernel_launch(void* const* d_in, const int* in_sizes, int n_in,
                              void* d_out, int out_size, void* d_ws, size_t ws_size,
                              hipStream_t stream) {
    (void)in_sizes; (void)n_in; (void)out_size; (void)ws_size;
    const float* pred = (const float*)d_in[0];
    const float* gt   = (const float*)d_in[1];
    const int*   idx  = (const int*)d_in[2];
    float* out      = (float*)d_out;
    float* partials = (float*)d_ws;      // (NG/TPB rounded up) * BATCH floats ~ 100 KB

    const int gpb = (NG + TPB - 1) / TPB;        // 391 blocks per image
    dim3 grid(gpb, BATCH);                       // 25024 blocks total
    vnl_partials_kernel<<<grid, TPB, 0, stream>>>(pred, gt, idx, partials);
    vnl_finalize_kernel<<<1, TPB, 0, stream>>>(partials, gpb * BATCH, out);
}